// SluiceTaskSharingUnit_3934190043240
// MI455X (gfx1250) — compile-verified
//
#include <hip/hip_runtime.h>

typedef float v2f __attribute__((ext_vector_type(2)));
typedef float v8f __attribute__((ext_vector_type(8)));

#define HIDDEN 1024
#define SUBSP  256          // HIDDEN / NUM_SUBSPACES
#define LRM    1.0f         // LR_MULT
#define ROWS   (4 * 4096)   // B * S

// One wave handles one (b,s) row half: 128 d-positions = 8 tiles of 16.
// Per tile: D(16x16) = A1(16x4:E^T k0..3) x B1(4x16: a-data)
//                    + A2(16x4:E^T k4..7) x B2(4x16: b-data)   [chained via C]
// D VGPR r, lanes 0..15 = out[j=r, pos n]  (rows 8..15 of A are zero -> lanes 16..31 unused).
__global__ __launch_bounds__(256)
void sluice_exchange_wmma(const float* __restrict__ Ain,
                          const float* __restrict__ Bin,
                          const float* __restrict__ Emat,
                          float* __restrict__ outA,
                          float* __restrict__ outB) {
  const int tid  = blockIdx.x * blockDim.x + threadIdx.x;
  const int wave = tid >> 5;
  const int lane = threadIdx.x & 31;
  const int bs   = wave >> 1;          // [0, 16384)
  const int d0   = (wave & 1) << 7;    // 0 or 128

  const int n  = lane & 15;            // D/B column = position-in-tile; also A-row m
  const int kh = lane >> 4;            // which K-half this lane's VGPRs carry

  // A-matrix VGPR layout (16x4 f32): lane m holds {K=2kh, K=2kh+1}.
  // A1[m,k] = E[k, m]      (k_global = 0..3, input_a subspaces)
  // A2[m,k] = E[4+k, m]    (k_global = 4..7, input_b subspaces)
  v2f a1, a2;
  if (n < 8) {
    a1.x = LRM * Emat[(2 * kh + 0) * 8 + n];
    a1.y = LRM * Emat[(2 * kh + 1) * 8 + n];
    a2.x = LRM * Emat[(2 * kh + 4) * 8 + n];
    a2.y = LRM * Emat[(2 * kh + 5) * 8 + n];
  } else {
    a1.x = 0.0f; a1.y = 0.0f; a2.x = 0.0f; a2.y = 0.0f;
  }

  const float* pa = Ain  + (size_t)bs * HIDDEN + d0;
  const float* pb = Bin  + (size_t)bs * HIDDEN + d0;
  float*       oa = outA + (size_t)bs * HIDDEN + d0;
  float*       ob = outB + (size_t)bs * HIDDEN + d0;

  // B-matrix (4x16 f32): VGPR0 = row K=2kh, VGPR1 = row K=2kh+1, column n.
  const int koff0 = (2 * kh + 0) * SUBSP + n;
  const int koff1 = (2 * kh + 1) * SUBSP + n;

#pragma unroll
  for (int t = 0; t < 8; ++t) {
    const int dt = t * 16;

    v2f b1, b2;
    b1.x = __builtin_nontemporal_load(pa + koff0 + dt);
    b1.y = __builtin_nontemporal_load(pa + koff1 + dt);
    b2.x = __builtin_nontemporal_load(pb + koff0 + dt);
    b2.y = __builtin_nontemporal_load(pb + koff1 + dt);

    v8f acc = {0.f, 0.f, 0.f, 0.f, 0.f, 0.f, 0.f, 0.f};
    // 8 args: (neg_a, A, neg_b, B, c_mod, C, reuse_a, reuse_b)
    acc = __builtin_amdgcn_wmma_f32_16x16x4_f32(false, a1, false, b1,
                                                (short)0, acc, false, false);
    acc = __builtin_amdgcn_wmma_f32_16x16x4_f32(false, a2, false, b2,
                                                (short)0, acc, false, false);

    if (kh == 0) {   // lanes 0..15 hold the valid D rows j=0..7
#pragma unroll
      for (int r = 0; r < 4; ++r)
        __builtin_nontemporal_store(acc[r], oa + r * SUBSP + dt + n);
#pragma unroll
      for (int r = 4; r < 8; ++r)
        __builtin_nontemporal_store(acc[r], ob + (r - 4) * SUBSP + dt + n);
    }
  }
}

extern "C" void kernel_launch(void* const* d_in, const int* in_sizes, int n_in,
                              void* d_out, int out_size, void* d_ws, size_t ws_size,
                              hipStream_t stream) {
  const float* a = (const float*)d_in[0];
  const float* b = (const float*)d_in[1];
  const float* E = (const float*)d_in[2];
  float* outA = (float*)d_out;
  float* outB = outA + (size_t)ROWS * HIDDEN;   // output_a flat, then output_b

  // 16384 rows * 2 waves/row = 32768 waves; 8 waves (256 thr) per block -> 4096 blocks
  dim3 grid(ROWS * 2 / 8);
  dim3 block(256);
  hipLaunchKernelGGL(sluice_exchange_wmma, grid, block, 0, stream,
                     a, b, E, outA, outB);
}